// SoftclDiceLoss_18872086298875
// MI455X (gfx1250) — compile-verified
//
#include <hip/hip_runtime.h>
#include <cstdint>

// Problem constants (B, C, D, H, W) = (2, 1, 64, 256, 256)
#define B_   2
#define D_   64
#define H_   256
#define W_   256
#define VOL  (D_ * H_ * W_)   // 4,194,304 per batch volume
#define N_   (B_ * VOL)       // 8,388,608 total elements per tensor
#define NV4  (N_ / 4)         // 2,097,152 float4 elements
#define TILE 16
#define LW   18               // 16 + 1 halo each side
#define PADV (-3.0e38f)

// Toggle for the gfx1250 async global->LDS path (inline asm).
#ifndef USE_ASYNC_LDS_ASM
#define USE_ASYNC_LDS_ASM 1
#endif

__device__ __forceinline__ float sigmoid_(float x) {
    return 1.0f / (1.0f + __expf(-x));
}

// -------------------------------------------------------------------------
// k_init: sp = sigmoid(logits)*mask ; st = target*mask ; zero the 4 sums.
// float4-vectorized: lowers to global_load_b128 / global_store_b128.
// -------------------------------------------------------------------------
__global__ void k_init(const float4* __restrict__ logits,
                       const float4* __restrict__ target,
                       const float4* __restrict__ mask,
                       float4* __restrict__ sp,
                       float4* __restrict__ st,
                       float* __restrict__ sums) {
    int i = blockIdx.x * blockDim.x + threadIdx.x;
    if (i == 0) {
        sums[0] = 0.0f; sums[1] = 0.0f; sums[2] = 0.0f; sums[3] = 0.0f;
    }
    if (i < NV4) {
        float4 l = logits[i];
        float4 t = target[i];
        float4 m = mask[i];
        float4 p, q;
        p.x = sigmoid_(l.x) * m.x;  q.x = t.x * m.x;
        p.y = sigmoid_(l.y) * m.y;  q.y = t.y * m.y;
        p.z = sigmoid_(l.z) * m.z;  q.z = t.z * m.z;
        p.w = sigmoid_(l.w) * m.w;  q.w = t.w * m.w;
        sp[i] = p;
        st[i] = q;
    }
}

// -------------------------------------------------------------------------
// k_crossmin: tmp = 7-point cross min of x (OOB taps skipped == +inf pad).
// One thread per (h,w) column; registers roll along D; prefetch next plane.
// -------------------------------------------------------------------------
__global__ void k_crossmin(const float* __restrict__ x,
                           float* __restrict__ tmp) {
    const int w = blockIdx.x * TILE + threadIdx.x;
    const int h = blockIdx.y * TILE + threadIdx.y;
    const int b = blockIdx.z;
    const size_t plane = (size_t)H_ * W_;
    const float* xb = x + (size_t)b * VOL;
    float* tb = tmp + (size_t)b * VOL;
    const size_t off = (size_t)h * W_ + w;

    float prv = 0.0f;
    float cur = xb[off];                 // d = 0
    float nxt = xb[plane + off];         // d = 1

    for (int d = 0; d < D_; ++d) {
        const size_t gi = (size_t)d * plane + off;
        float mn = cur;
        if (d > 0)       mn = fminf(mn, prv);
        if (d < D_ - 1)  mn = fminf(mn, nxt);
        if (h > 0)       mn = fminf(mn, xb[gi - W_]);
        if (h < H_ - 1)  mn = fminf(mn, xb[gi + W_]);
        if (w > 0)       mn = fminf(mn, xb[gi - 1]);
        if (w < W_ - 1)  mn = fminf(mn, xb[gi + 1]);
        tb[gi] = mn;

        prv = cur;
        cur = nxt;
        if (d + 2 < D_) {
            nxt = xb[gi + 2 * plane];
            if (d + 3 < D_)
                __builtin_prefetch(xb + gi + 3 * plane, 0, 0);  // global_prefetch_b8
        } else {
            nxt = 0.0f;
        }
    }
}

// -------------------------------------------------------------------------
// LDS plane loader for k_maxupdate: fill an 18x18 halo tile of plane d with
// -inf prefill, then overwrite in-range elements via async global->LDS loads
// (gfx1250 global_load_async_to_lds_b32, drained by s_wait_asynccnt).
// -------------------------------------------------------------------------
__device__ __forceinline__ void load_plane(float* __restrict__ dst,        // LDS, 18*18
                                           const float* __restrict__ srcb, // tmp + b*VOL
                                           int d, int h0, int w0) {
    const int lin = threadIdx.y * TILE + threadIdx.x;  // 0..255
    __syncthreads();  // previous consumers of this slot are done
    for (int i = lin; i < LW * LW; i += TILE * TILE)
        dst[i] = PADV;
    __syncthreads();  // prefill visible before async writes land

    if (d >= 0 && d < D_) {
        const float* pl = srcb + (size_t)d * H_ * W_;
        for (int i = lin; i < LW * LW; i += TILE * TILE) {
            const int lh = i / LW, lw = i % LW;
            const int h = h0 - 1 + lh;
            const int w = w0 - 1 + lw;
            if ((unsigned)h < (unsigned)H_ && (unsigned)w < (unsigned)W_) {
#if USE_ASYNC_LDS_ASM
                unsigned lds_off = (unsigned)(uintptr_t)(&dst[i]);
                unsigned long long gaddr =
                    (unsigned long long)(uintptr_t)(pl + (size_t)h * W_ + w);
                asm volatile("global_load_async_to_lds_b32 %0, %1, off"
                             :
                             : "v"(lds_off), "v"(gaddr)
                             : "memory");
#else
                dst[i] = pl[(size_t)h * W_ + w];
#endif
            }
        }
    }
#if USE_ASYNC_LDS_ASM
    asm volatile("s_wait_asynccnt 0x0" ::: "memory");
#endif
    __syncthreads();
}

// -------------------------------------------------------------------------
// k_maxupdate: mp = 3x3x3 max of tmp (pad -inf); x = relu(x - relu(mp - tmp))
// 3 rolling 18x18 halo planes in LDS. Max over slots is order-independent,
// only the center tap needs the slot currently holding plane d.
// -------------------------------------------------------------------------
__global__ void k_maxupdate(float* __restrict__ x,
                            const float* __restrict__ tmp) {
    __shared__ float pl[3][LW * LW];
    const int tx = threadIdx.x, ty = threadIdx.y;
    const int w0 = blockIdx.x * TILE, h0 = blockIdx.y * TILE;
    const int b = blockIdx.z;
    const float* tb = tmp + (size_t)b * VOL;
    float* xb = x + (size_t)b * VOL;

    load_plane(pl[0], tb, -1, h0, w0);
    load_plane(pl[1], tb,  0, h0, w0);
    load_plane(pl[2], tb,  1, h0, w0);
    int sc = 1;  // slot holding plane d

    for (int d = 0; d < D_; ++d) {
        float mp = PADV;
#pragma unroll
        for (int s = 0; s < 3; ++s)
#pragma unroll
            for (int dh = 0; dh < 3; ++dh)
#pragma unroll
                for (int dw = 0; dw < 3; ++dw)
                    mp = fmaxf(mp, pl[s][(ty + dh) * LW + (tx + dw)]);

        const float center = pl[sc][(ty + 1) * LW + (tx + 1)];
        const float contour = fmaxf(mp - center, 0.0f);
        const size_t gi = (size_t)d * H_ * W_ + (size_t)(h0 + ty) * W_ + (w0 + tx);
        xb[gi] = fmaxf(xb[gi] - contour, 0.0f);

        const int sold = (sc + 2) % 3;           // slot holding plane d-1
        load_plane(pl[sold], tb, d + 2, h0, w0); // becomes plane d+2 (has barriers)
        sc = (sc + 1) % 3;
    }
}

// -------------------------------------------------------------------------
// k_reduce: S0=sum(st*pred) S1=sum(st) S2=sum(sp*tgt) S3=sum(sp)
// pred/tgt recomputed on the fly from inputs (saves 67 MB of workspace).
// float4-vectorized grid-stride loop; wave32 shuffle reduction; f32 atomics.
// -------------------------------------------------------------------------
__global__ void k_reduce(const float4* __restrict__ logits,
                         const float4* __restrict__ target,
                         const float4* __restrict__ mask,
                         const float4* __restrict__ sp,
                         const float4* __restrict__ st,
                         float* __restrict__ sums) {
    float s0 = 0.f, s1 = 0.f, s2 = 0.f, s3 = 0.f;
    const size_t stride = (size_t)gridDim.x * blockDim.x;
    for (size_t i = (size_t)blockIdx.x * blockDim.x + threadIdx.x; i < NV4; i += stride) {
        const float4 l = logits[i];
        const float4 t = target[i];
        const float4 m = mask[i];
        const float4 a = st[i];
        const float4 c = sp[i];
        // lane .x
        float pr = sigmoid_(l.x) * m.x, tg = t.x * m.x;
        s0 += a.x * pr; s1 += a.x; s2 += c.x * tg; s3 += c.x;
        // lane .y
        pr = sigmoid_(l.y) * m.y; tg = t.y * m.y;
        s0 += a.y * pr; s1 += a.y; s2 += c.y * tg; s3 += c.y;
        // lane .z
        pr = sigmoid_(l.z) * m.z; tg = t.z * m.z;
        s0 += a.z * pr; s1 += a.z; s2 += c.z * tg; s3 += c.z;
        // lane .w
        pr = sigmoid_(l.w) * m.w; tg = t.w * m.w;
        s0 += a.w * pr; s1 += a.w; s2 += c.w * tg; s3 += c.w;
    }
    // wave32 reduction
#pragma unroll
    for (int o = 16; o > 0; o >>= 1) {
        s0 += __shfl_down(s0, o, 32);
        s1 += __shfl_down(s1, o, 32);
        s2 += __shfl_down(s2, o, 32);
        s3 += __shfl_down(s3, o, 32);
    }
    __shared__ float sh[4][8];  // 256 threads / wave32 = 8 waves
    const int lane = threadIdx.x & 31, wv = threadIdx.x >> 5;
    if (lane == 0) { sh[0][wv] = s0; sh[1][wv] = s1; sh[2][wv] = s2; sh[3][wv] = s3; }
    __syncthreads();
    if (threadIdx.x == 0) {
        float t0 = 0.f, t1 = 0.f, t2 = 0.f, t3 = 0.f;
#pragma unroll
        for (int k = 0; k < 8; ++k) { t0 += sh[0][k]; t1 += sh[1][k]; t2 += sh[2][k]; t3 += sh[3][k]; }
        atomicAdd(&sums[0], t0);
        atomicAdd(&sums[1], t1);
        atomicAdd(&sums[2], t2);
        atomicAdd(&sums[3], t3);
    }
}

__global__ void k_final(const float* __restrict__ sums, float* __restrict__ out) {
    const float r = (sums[0] + 1e-12f) / (sums[1] + 1e-12f);  // clrecall
    const float a = (sums[2] + 1e-12f) / (sums[3] + 1e-12f);  // clacc
    out[0] = 1.0f - 2.0f * r * a / (r + a + 1e-12f);
}

// -------------------------------------------------------------------------
// kernel_launch
// workspace layout (floats): [0,N): sp  [N,2N): st  [2N,3N): tmp  [3N,3N+4): sums
// total = 3*N_*4 + 16 bytes ~= 100.7 MB  (fits in the 192 MB L2 -> the whole
// 5-iteration erosion loop runs at L2 bandwidth, not HBM)
// -------------------------------------------------------------------------
extern "C" void kernel_launch(void* const* d_in, const int* in_sizes, int n_in,
                              void* d_out, int out_size, void* d_ws, size_t ws_size,
                              hipStream_t stream) {
    const float* logits = (const float*)d_in[0];
    const float* target = (const float*)d_in[1];
    const float* mask   = (const float*)d_in[2];
    float* ws   = (float*)d_ws;
    float* sp   = ws;
    float* st   = ws + (size_t)N_;
    float* tmp  = ws + 2 * (size_t)N_;
    float* sums = ws + 3 * (size_t)N_;
    float* out  = (float*)d_out;

    k_init<<<(NV4 + 255) / 256, 256, 0, stream>>>(
        (const float4*)logits, (const float4*)target, (const float4*)mask,
        (float4*)sp, (float4*)st, sums);

    dim3 blk(TILE, TILE, 1);
    dim3 grd(W_ / TILE, H_ / TILE, B_);
    for (int it = 0; it < 5; ++it) {
        for (int t = 0; t < 2; ++t) {
            float* xbuf = t ? sp : st;
            k_crossmin<<<grd, blk, 0, stream>>>(xbuf, tmp);
            k_maxupdate<<<grd, blk, 0, stream>>>(xbuf, tmp);
        }
    }

    k_reduce<<<2048, 256, 0, stream>>>(
        (const float4*)logits, (const float4*)target, (const float4*)mask,
        (const float4*)sp, (const float4*)st, sums);
    k_final<<<1, 1, 0, stream>>>(sums, out);
}